// GCNEncoder_43688407335391
// MI455X (gfx1250) — compile-verified
//
#include <hip/hip_runtime.h>
#include <hip/hip_bf16.h>

typedef __attribute__((ext_vector_type(2))) float v2f;
typedef __attribute__((ext_vector_type(4))) float v4f;
typedef __attribute__((ext_vector_type(8))) float v8f;

#ifndef THREADS
#define THREADS 256
#endif

// ---------------------------------------------------------------------------
// Edge-weight pipeline
// ---------------------------------------------------------------------------
__global__ void k_zero(float* __restrict__ p, int n) {
  int i = blockIdx.x * blockDim.x + threadIdx.x;
  if (i < n) p[i] = 0.0f;
}

__global__ void k_deg(const int* __restrict__ row, float* __restrict__ deg, int nE) {
  int i = blockIdx.x * blockDim.x + threadIdx.x;
  if (i < nE) atomicAdd(&deg[row[i]], 1.0f);
}

__global__ void k_dinv(const float* __restrict__ deg, float* __restrict__ dinv, int n) {
  int i = blockIdx.x * blockDim.x + threadIdx.x;
  if (i < n) {
    float d = deg[i];
    dinv[i] = (d > 0.0f) ? rsqrtf(fmaxf(d, 1.0f)) : 0.0f;
  }
}

__global__ void k_edgew(const int* __restrict__ row, const int* __restrict__ col,
                        const float* __restrict__ dinv, float* __restrict__ w, int nE) {
  int i = blockIdx.x * blockDim.x + threadIdx.x;
  if (i < nE) w[i] = -dinv[row[i]] * dinv[col[i]];
}

// ---------------------------------------------------------------------------
// Wcat = [W0;W1;W2] (Kpad x Cout, zero-padded rows)
// ---------------------------------------------------------------------------
__global__ void k_build_wcat(const float* __restrict__ W0, const float* __restrict__ W1,
                             const float* __restrict__ W2, float* __restrict__ Wcat,
                             int Cin, int Cout, int Kpad) {
  int idx = blockIdx.x * blockDim.x + threadIdx.x;
  int total = Kpad * Cout;
  if (idx >= total) return;
  int r = idx / Cout;
  int c = idx - r * Cout;
  float v = 0.0f;
  if (r < Cin)            v = W0[r * Cout + c];
  else if (r < 2 * Cin)   v = W1[(r - Cin) * Cout + c];
  else if (r < 3 * Cin)   v = W2[(r - 2 * Cin) * Cout + c];
  Wcat[idx] = v;
}

// Xcat = [x | 0 | -x | padzeros]  (M x Kpad); z2 region seeded with -x so the
// second SpMM accumulates 2*w*z1[col] straight into z2 = 2*lhat(z1) - x.
__global__ void k_init_xcat(const float* __restrict__ x, float* __restrict__ Xcat,
                            int Cin, int Kpad, int M) {
  int idx = blockIdx.x * blockDim.x + threadIdx.x;
  int total = M * Kpad;
  if (idx >= total) return;
  int m = idx / Kpad;
  int j = idx - m * Kpad;
  float v = 0.0f;
  if (j < Cin)                          v = x[(size_t)m * Cin + j];
  else if (j >= 2 * Cin && j < 3 * Cin) v = -x[(size_t)m * Cin + (j - 2 * Cin)];
  Xcat[idx] = v;
}

// ---------------------------------------------------------------------------
// SpMM (float2/thread): Xcat[row, dstOff+2c..] += scale*w[e]*Xcat[col, srcOff+2c..]
// All Cin are even. Gathers are b64; atomics are HW fp32 adds into L2.
// ---------------------------------------------------------------------------
__global__ void k_spmm(const int* __restrict__ row, const int* __restrict__ col,
                       const float* __restrict__ w, float* __restrict__ Xcat,
                       int Kpad, int srcOff, int dstOff, int halfC, float scale, int nE) {
  int idx = blockIdx.x * blockDim.x + threadIdx.x;
  int total = nE * halfC;
  if (idx >= total) return;
  int e  = idx / halfC;
  int ch = (idx - e * halfC) * 2;
  v2f xv = *(const v2f*)(Xcat + (size_t)col[e] * Kpad + srcOff + ch);
  float s = scale * w[e];
  float* dst = Xcat + (size_t)row[e] * Kpad + dstOff + ch;
  atomicAdd(dst + 0, s * xv.x);
  atomicAdd(dst + 1, s * xv.y);
}

// ---------------------------------------------------------------------------
// WMMA GEMM (fp32): out = relu(Xcat[M,Kpad] @ Wcat[Kpad,N] + bias)
// Block = 8 waves; block tile 256(M) x 64(N); wave tile 32(M) x 64(N) with
// 2x4 accumulators: each B fragment (ds_load_2addr) feeds 2 WMMAs, each A
// fragment feeds 4. W staged through LDS in 32-deep K chunks, software-
// pipelined (next chunk loads to regs while computing current).
// LDS row stride 72 -> the two half-wave row reads hit disjoint banks.
// wave32 fragment layouts (CDNA5 ISA 7.12.2):
//   A 16x4 : contiguous float2 at xrow[k + 2*half] (K halves across lane halves)
//   B 4x16 : rows striped across lanes; K halves across lane halves
//   C/D    : VGPR r -> row r (lanes 0-15) / row r+8 (lanes 16-31), N = lane%16
// ---------------------------------------------------------------------------
#define KC 32
#define LDS_STRIDE 72

__global__ __launch_bounds__(256)
void k_gemm_wmma(const float* __restrict__ Xcat, const float* __restrict__ Wcat,
                 const float* __restrict__ bias, float* __restrict__ out,
                 int M, int Kpad, int N) {
  __shared__ float sW[KC][LDS_STRIDE];

  const int tid  = threadIdx.x;
  const int wave = tid >> 5;
  const int lane = tid & 31;
  const int half = lane >> 4;
  const int lm   = lane & 15;

  const int tileM  = (blockIdx.x * 8 + wave) * 32;
  const int tileN  = blockIdx.y * 64;
  const int tileMc = tileM < M ? tileM : (M - 32);   // clamp; store guarded later

  // Cooperative W staging: each thread owns 8 consecutive floats of one row.
  const int sr = tid >> 3;          // 0..31 : row within chunk
  const int sc = (tid & 7) * 8;     // 0..56 : col within 64-wide tile
  const float* wsrc = Wcat + (size_t)sr * N + tileN + sc;

  const float* __restrict__ xrow0 = Xcat + (size_t)(tileMc + lm) * Kpad;
  const float* __restrict__ xrow1 = xrow0 + (size_t)16 * Kpad;

  v8f acc[2][4];
#pragma unroll
  for (int s = 0; s < 2; ++s)
#pragma unroll
    for (int t = 0; t < 4; ++t) acc[s][t] = (v8f){};

  const int nChunks = Kpad / KC;
  v4f ra = *(const v4f*)(wsrc);
  v4f rb = *(const v4f*)(wsrc + 4);

  for (int ch = 0; ch < nChunks; ++ch) {
    *(v4f*)&sW[sr][sc]     = ra;
    *(v4f*)&sW[sr][sc + 4] = rb;
    __syncthreads();

    if (ch + 1 < nChunks) {
      wsrc += (size_t)KC * N;
      ra = *(const v4f*)(wsrc);
      rb = *(const v4f*)(wsrc + 4);
    }

    const float* xk0 = xrow0 + ch * KC;
    const float* xk1 = xrow1 + ch * KC;
#pragma unroll
    for (int kk = 0; kk < KC; kk += 4) {
      const int r0 = kk + 2 * half;
      v2f a0 = *(const v2f*)(xk0 + kk + 2 * half);
      v2f a1 = *(const v2f*)(xk1 + kk + 2 * half);
      v2f b0, b1, b2, b3;
      b0.x = sW[r0][lm];      b0.y = sW[r0 + 1][lm];
      b1.x = sW[r0][16 + lm]; b1.y = sW[r0 + 1][16 + lm];
      b2.x = sW[r0][32 + lm]; b2.y = sW[r0 + 1][32 + lm];
      b3.x = sW[r0][48 + lm]; b3.y = sW[r0 + 1][48 + lm];
      acc[0][0] = __builtin_amdgcn_wmma_f32_16x16x4_f32(false, a0, false, b0, (short)0, acc[0][0], false, false);
      acc[1][0] = __builtin_amdgcn_wmma_f32_16x16x4_f32(false, a1, false, b0, (short)0, acc[1][0], false, false);
      acc[0][1] = __builtin_amdgcn_wmma_f32_16x16x4_f32(false, a0, false, b1, (short)0, acc[0][1], false, false);
      acc[1][1] = __builtin_amdgcn_wmma_f32_16x16x4_f32(false, a1, false, b1, (short)0, acc[1][1], false, false);
      acc[0][2] = __builtin_amdgcn_wmma_f32_16x16x4_f32(false, a0, false, b2, (short)0, acc[0][2], false, false);
      acc[1][2] = __builtin_amdgcn_wmma_f32_16x16x4_f32(false, a1, false, b2, (short)0, acc[1][2], false, false);
      acc[0][3] = __builtin_amdgcn_wmma_f32_16x16x4_f32(false, a0, false, b3, (short)0, acc[0][3], false, false);
      acc[1][3] = __builtin_amdgcn_wmma_f32_16x16x4_f32(false, a1, false, b3, (short)0, acc[1][3], false, false);
    }
    __syncthreads();
  }

  if (tileM < M) {
    const float bv0 = bias[tileN + lm];
    const float bv1 = bias[tileN + 16 + lm];
    const float bv2 = bias[tileN + 32 + lm];
    const float bv3 = bias[tileN + 48 + lm];
#pragma unroll
    for (int s = 0; s < 2; ++s) {
      float* orow = out + (size_t)(tileM + s * 16 + 8 * half) * N + tileN + lm;
#pragma unroll
      for (int r = 0; r < 8; ++r) {
        float* p = orow + (size_t)r * N;
        float v0 = acc[s][0][r] + bv0, v1 = acc[s][1][r] + bv1;
        float v2 = acc[s][2][r] + bv2, v3 = acc[s][3][r] + bv3;
        p[0]  = v0 > 0.0f ? v0 : 0.0f;
        p[16] = v1 > 0.0f ? v1 : 0.0f;
        p[32] = v2 > 0.0f ? v2 : 0.0f;
        p[48] = v3 > 0.0f ? v3 : 0.0f;
      }
    }
  }
}

// ---------------------------------------------------------------------------
// Launch
// ---------------------------------------------------------------------------
extern "C" void kernel_launch(void* const* d_in, const int* in_sizes, int n_in,
                              void* d_out, int out_size, void* d_ws, size_t ws_size,
                              hipStream_t stream) {
  (void)n_in; (void)out_size; (void)ws_size;

  const float* v     = (const float*)d_in[0];
  const int*   edges = (const int*)d_in[1];
  const int nE = in_sizes[1] / 2;
  const int nN = in_sizes[0] / 86;
  const int* row = edges;
  const int* col = edges + nE;

  const float* W[3][3];
  const float* B[3];
  for (int l = 0; l < 3; ++l) {
    W[l][0] = (const float*)d_in[2 + l * 4 + 0];
    W[l][1] = (const float*)d_in[2 + l * 4 + 1];
    W[l][2] = (const float*)d_in[2 + l * 4 + 2];
    B[l]    = (const float*)d_in[2 + l * 4 + 3];
  }

  // Workspace layout (floats): deg | dinv | w | Wcat(768*512) | Xcat(nN*768)
  float* ws   = (float*)d_ws;
  float* deg  = ws;  ws += nN;
  float* dinv = ws;  ws += nN;
  float* ew   = ws;  ws += nE;
  float* Wcat = ws;  ws += 768 * 512;
  float* Xcat = ws;

  const int T = THREADS;
  k_zero <<<(nN + T - 1) / T, T, 0, stream>>>(deg, nN);
  k_deg  <<<(nE + T - 1) / T, T, 0, stream>>>(row, deg, nE);
  k_dinv <<<(nN + T - 1) / T, T, 0, stream>>>(deg, dinv, nN);
  k_edgew<<<(nE + T - 1) / T, T, 0, stream>>>(row, col, dinv, ew, nE);

  float* out1 = (float*)d_out;
  float* out2 = out1 + (size_t)nN * 128;
  float* out3 = out2 + (size_t)nN * 256;
  float* outs[3] = {out1, out2, out3};
  const int dims[4] = {86, 128, 256, 512};

  const float* x = v;
  for (int l = 0; l < 3; ++l) {
    const int Cin  = dims[l];
    const int Cout = dims[l + 1];
    const int Kpad = (3 * Cin + 31) & ~31;   // 288 / 384 / 768 (multiple of KC)

    int twc = Kpad * Cout;
    k_build_wcat<<<(twc + T - 1) / T, T, 0, stream>>>(W[l][0], W[l][1], W[l][2],
                                                      Wcat, Cin, Cout, Kpad);
    int txc = nN * Kpad;
    k_init_xcat<<<(txc + T - 1) / T, T, 0, stream>>>(x, Xcat, Cin, Kpad, nN);

    int ts = nE * (Cin / 2);
    // z1 = lhat(x)
    k_spmm<<<(ts + T - 1) / T, T, 0, stream>>>(row, col, ew, Xcat, Kpad,
                                               0, Cin, Cin / 2, 1.0f, nE);
    // z2 = 2*lhat(z1) - x   (z2 region pre-seeded with -x)
    k_spmm<<<(ts + T - 1) / T, T, 0, stream>>>(row, col, ew, Xcat, Kpad,
                                               Cin, 2 * Cin, Cin / 2, 2.0f, nE);

    dim3 grid((nN / 32 + 7) / 8, Cout / 64);
    k_gemm_wmma<<<grid, 256, 0, stream>>>(Xcat, Wcat, B[l], outs[l], nN, Kpad, Cout);

    x = outs[l];
  }
}